// AttentionWithGeGLU_33526514712618
// MI455X (gfx1250) — compile-verified
//
#include <hip/hip_runtime.h>
#include <math.h>

// Problem dims (AttentionWithGeGLU): B=32, S=2048, D=1024, OUT=4096
#define BB   32
#define SS   2048
#define DD   1024
#define OUTN 4096
#define H2   (2 * OUTN)   // 8192
#define EPSV 1e-6f

typedef __attribute__((ext_vector_type(2))) float v2f;
typedef __attribute__((ext_vector_type(4))) float v4f;
typedef __attribute__((ext_vector_type(8))) float v8f;

// ---------------------------------------------------------------------------
// Kernel 1: fused RMS-stats + attention score. One wave32 per (b,s) row.
// score[row] = rsqrt(mean(x^2)+eps) * sum_d(x * ln_w * att_w) + att_b
// Also store rstd[row] for later use in pooled accumulation.
// Single pass over the 256 MiB x tensor (coalesced float4 loads).
// ---------------------------------------------------------------------------
__global__ __launch_bounds__(256) void k_rms_score(
    const float* __restrict__ x, const float* __restrict__ ln_w,
    const float* __restrict__ att_w, const float* __restrict__ att_b,
    float* __restrict__ score, float* __restrict__ rstd) {
  const int lane = threadIdx.x & 31;
  const int wave = threadIdx.x >> 5;
  const int row  = blockIdx.x * 8 + wave;      // row in [0, B*S)
  const float* xr = x + (size_t)row * DD;

  float ss = 0.f, sv = 0.f;
#pragma unroll
  for (int i = 0; i < 8; ++i) {
    const int idx = (i * 32 + lane) * 4;       // contiguous across the wave
    v4f xv = *(const v4f*)(xr + idx);
    v4f lw = *(const v4f*)(ln_w + idx);
    v4f aw = *(const v4f*)(att_w + idx);
    ss += xv.x * xv.x + xv.y * xv.y + xv.z * xv.z + xv.w * xv.w;
    sv += xv.x * lw.x * aw.x + xv.y * lw.y * aw.y +
          xv.z * lw.z * aw.z + xv.w * lw.w * aw.w;
  }
  // wave32 butterfly reduction
#pragma unroll
  for (int off = 16; off >= 1; off >>= 1) {
    ss += __shfl_xor(ss, off, 32);
    sv += __shfl_xor(sv, off, 32);
  }
  if (lane == 0) {
    const float r = rsqrtf(ss * (1.0f / DD) + EPSV);
    rstd[row]  = r;
    score[row] = sv * r + att_b[0];
  }
}

// ---------------------------------------------------------------------------
// Kernel 2: softmax over S per batch; fold rstd into the coefficient:
// coef[b,s] = softmax_s(score)[b,s] * rstd[b,s]
// One 256-thread block per batch (8 scores/thread).
// ---------------------------------------------------------------------------
__global__ __launch_bounds__(256) void k_softmax_coef(
    const float* __restrict__ score, const float* __restrict__ rstd,
    float* __restrict__ coef) {
  __shared__ float red[256];
  const int b   = blockIdx.x;
  const int tid = threadIdx.x;
  const float* sc = score + b * SS;

  float loc[8];
  float mx = -3.0e38f;
#pragma unroll
  for (int i = 0; i < 8; ++i) {
    loc[i] = sc[tid + 256 * i];
    mx = fmaxf(mx, loc[i]);
  }
  red[tid] = mx; __syncthreads();
  for (int st = 128; st >= 1; st >>= 1) {
    if (tid < st) red[tid] = fmaxf(red[tid], red[tid + st]);
    __syncthreads();
  }
  mx = red[0]; __syncthreads();

  float sum = 0.f;
#pragma unroll
  for (int i = 0; i < 8; ++i) {
    loc[i] = expf(loc[i] - mx);
    sum += loc[i];
  }
  red[tid] = sum; __syncthreads();
  for (int st = 128; st >= 1; st >>= 1) {
    if (tid < st) red[tid] += red[tid + st];
    __syncthreads();
  }
  const float inv = 1.0f / red[0];
#pragma unroll
  for (int i = 0; i < 8; ++i) {
    const int s = tid + 256 * i;
    coef[b * SS + s] = loc[i] * inv * rstd[b * SS + s];
  }
}

// ---------------------------------------------------------------------------
// Kernel 3: pooled partials. partial[chunk,b,d] = sum_{s in chunk} coef*x
// Grid (D/256, S-chunks=8, B). Coalesced reads of x (threads span d),
// coefficients staged in LDS. Deterministic (no float atomics).
// ---------------------------------------------------------------------------
__global__ __launch_bounds__(256) void k_pooled_partial(
    const float* __restrict__ x, const float* __restrict__ coef,
    float* __restrict__ partial) {
  __shared__ float cs[256];
  const int d  = blockIdx.x * 256 + threadIdx.x;
  const int sc = blockIdx.y;                    // s-chunk 0..7
  const int b  = blockIdx.z;
  cs[threadIdx.x] = coef[b * SS + sc * 256 + threadIdx.x];
  __syncthreads();

  const float* xb = x + ((size_t)b * SS + (size_t)sc * 256) * DD + d;
  float acc = 0.f;
#pragma unroll 4
  for (int s = 0; s < 256; ++s) acc += cs[s] * xb[(size_t)s * DD];
  partial[((sc * BB + b) * DD) + d] = acc;
}

// ---------------------------------------------------------------------------
// Kernel 4: reduce 8 partials, apply ln_w: pooled[b,d]
// ---------------------------------------------------------------------------
__global__ __launch_bounds__(256) void k_reduce_pooled(
    const float* __restrict__ partial, const float* __restrict__ ln_w,
    float* __restrict__ pooled) {
  const int i = blockIdx.x * 256 + threadIdx.x;   // [0, B*D)
  const int d = i & (DD - 1);
  float acc = 0.f;
#pragma unroll
  for (int c = 0; c < 8; ++c) acc += partial[c * BB * DD + i];
  pooled[i] = acc * ln_w[d];
}

// ---------------------------------------------------------------------------
// Kernel 5: WMMA GEMM  h[32,8192] = pooled[32,1024] @ geglu_w[1024,8192] + b
// V_WMMA_F32_16X16X4_F32 (f32-exact). One wave owns a 32x16 output tile
// (two 16x16 f32 accumulators), K loop in steps of 4.
// ISA layouts (CDNA5 ISA):
//   A 16x4:  lanes 0-15 -> K={0,1}, lanes 16-31 -> K={2,3} (2 VGPRs)
//   B 4x16:  row K = vgpr + 2*(lane>=16), N = lane%16      (2 VGPRs)
//   C/D 16x16: row M = vgpr + 8*(lane>=16), N = lane%16    (8 VGPRs)
// 64 blocks x 8 waves cover N=8192 -> geglu_w streamed exactly once (32 MiB).
// ---------------------------------------------------------------------------
__global__ __launch_bounds__(256) void k_gemm_wmma(
    const float* __restrict__ pooled, const float* __restrict__ gw,
    const float* __restrict__ gb, float* __restrict__ h) {
  const int lane = threadIdx.x & 31;
  const int wave = threadIdx.x >> 5;
  const int tn   = blockIdx.x * 8 + wave;       // N-tile in [0, 512)
  const int col  = tn * 16 + (lane & 15);       // global column
  const int koff = (lane >> 4) * 2;             // K sub-offset per half-wave
  const int mrow = (lane & 15);                 // A row for this lane

  v8f acc0 = {};   // rows 0..15
  v8f acc1 = {};   // rows 16..31

  for (int k0 = 0; k0 < DD; k0 += 4) {
    v2f a0 = *(const v2f*)(pooled + (size_t)mrow * DD + k0 + koff);
    v2f a1 = *(const v2f*)(pooled + (size_t)(mrow + 16) * DD + k0 + koff);
    v2f bb;
    bb.x = gw[(size_t)(k0 + koff) * H2 + col];
    bb.y = gw[(size_t)(k0 + koff + 1) * H2 + col];
    acc0 = __builtin_amdgcn_wmma_f32_16x16x4_f32(
        false, a0, false, bb, (short)0, acc0, false, false);
    acc1 = __builtin_amdgcn_wmma_f32_16x16x4_f32(
        false, a1, false, bb, (short)0, acc1, false, false);
  }

  const int rbase = 8 * (lane >> 4);
  const float bias = gb[col];
#pragma unroll
  for (int r = 0; r < 8; ++r) {
    const int m = r + rbase;
    h[(size_t)m * H2 + col]        = acc0[r] + bias;
    h[(size_t)(m + 16) * H2 + col] = acc1[r] + bias;
  }
}

// ---------------------------------------------------------------------------
// Kernel 6: GeGLU epilogue out[b,j] = val * 0.5*g*(1+erf(g/sqrt(2)))
// ---------------------------------------------------------------------------
__global__ __launch_bounds__(256) void k_geglu(
    const float* __restrict__ h, float* __restrict__ out) {
  const int i = blockIdx.x * 256 + threadIdx.x;   // [0, B*OUT)
  const int b = i >> 12;                          // / 4096
  const int j = i & (OUTN - 1);
  const float val  = h[(size_t)b * H2 + j];
  const float gate = h[(size_t)b * H2 + OUTN + j];
  const float g = 0.5f * gate * (1.0f + erff(gate * 0.70710678118654752f));
  out[i] = val * g;
}

// ---------------------------------------------------------------------------
// Launcher
// ---------------------------------------------------------------------------
extern "C" void kernel_launch(void* const* d_in, const int* in_sizes, int n_in,
                              void* d_out, int out_size, void* d_ws,
                              size_t ws_size, hipStream_t stream) {
  (void)in_sizes; (void)n_in; (void)out_size; (void)ws_size;
  const float* x     = (const float*)d_in[0];   // [B,S,D]
  const float* ln_w  = (const float*)d_in[1];   // [D]
  const float* att_w = (const float*)d_in[2];   // [D,1]
  const float* att_b = (const float*)d_in[3];   // [1]
  const float* gw    = (const float*)d_in[4];   // [D, 2*OUT]
  const float* gb    = (const float*)d_in[5];   // [2*OUT]
  float* out = (float*)d_out;                   // [B, OUT]

  // Workspace layout (floats): every region fully written before read.
  float* ws      = (float*)d_ws;
  float* score   = ws;                          // B*S        = 65536
  float* rstd    = ws + 65536;                  // B*S        = 65536
  float* coef    = ws + 131072;                 // B*S        = 65536
  float* partial = ws + 196608;                 // 8*B*D      = 262144
  float* pooled  = ws + 458752;                 // B*D        = 32768
  float* hbuf    = ws + 491520;                 // B*2*OUT    = 262144

  k_rms_score<<<(BB * SS) / 8, 256, 0, stream>>>(x, ln_w, att_w, att_b,
                                                 score, rstd);
  k_softmax_coef<<<BB, 256, 0, stream>>>(score, rstd, coef);
  k_pooled_partial<<<dim3(DD / 256, 8, BB), 256, 0, stream>>>(x, coef, partial);
  k_reduce_pooled<<<(BB * DD) / 256, 256, 0, stream>>>(partial, ln_w, pooled);
  k_gemm_wmma<<<(H2 / 16) / 8, 256, 0, stream>>>(pooled, gw, gb, hbuf);
  k_geglu<<<(BB * OUTN) / 256, 256, 0, stream>>>(hbuf, out);
}